// HoleFiller_85341000171679
// MI455X (gfx1250) — compile-verified
//
#include <hip/hip_runtime.h>
#include <stdint.h>

// Problem constants (match reference)
#define BATCH   8
#define IMG_H   1024
#define IMG_W   1024
#define ITERS   50
#define RAD     3          // (7-1)/2

// Tiling
#define TILE_H  32
#define TILE_W  64
#define RAW_H   (TILE_H + 2*RAD)   // 38
#define RAW_W   (TILE_W + 2*RAD)   // 70
#define RAW_P   72                 // padded pitch
#define HS_P    TILE_W             // 64
#define HS_ROWS (RAW_H + 2)        // 40 (2 zero pad rows for K-padding to 24)

typedef __attribute__((ext_vector_type(2))) float v2f;
typedef __attribute__((ext_vector_type(8))) float v8f;

#if defined(__has_builtin)
#if __has_builtin(__builtin_amdgcn_global_load_async_to_lds_b32) && \
    __has_builtin(__builtin_amdgcn_s_wait_asynccnt)
#define USE_ASYNC 1
#pragma message("CDNA5 path: global_load_async_to_lds_b32 ENABLED")
#else
#pragma message("CDNA5 path: async-to-LDS builtin NOT available, plain loads")
#endif
#if __has_builtin(__builtin_amdgcn_wmma_f32_16x16x4_f32)
#define USE_WMMA 1
#pragma message("CDNA5 path: v_wmma_f32_16x16x4_f32 ENABLED")
#else
#pragma message("CDNA5 path: wmma f32 16x16x4 builtin NOT available, VALU fallback")
#endif
#endif

__global__ __launch_bounds__(256)
void fill_iter_kernel(const float* __restrict__ src, float* __restrict__ dst) {
    __shared__ float raw[RAW_H * RAW_P];     // tile + halo of "filled"
    __shared__ float hsD[HS_ROWS * HS_P];    // horizontal 7-sum of data
    __shared__ float hsM[HS_ROWS * HS_P];    // horizontal 7-sum of mask

    const int tid = threadIdx.x;
    const int gx0 = blockIdx.x * TILE_W;
    const int gy0 = blockIdx.y * TILE_H;
    const float* plane  = src + (size_t)blockIdx.z * IMG_H * IMG_W;
    float*       oplane = dst + (size_t)blockIdx.z * IMG_H * IMG_W;

    // ---- phase 0: zero raw tile (image-border zero pad) + hsum K-pad rows ----
    for (int i = tid; i < RAW_H * RAW_P; i += 256) raw[i] = 0.0f;
    for (int i = tid; i < 2 * HS_P; i += 256) {
        hsD[RAW_H * HS_P + i] = 0.0f;
        hsM[RAW_H * HS_P + i] = 0.0f;
    }
    __syncthreads();

    // ---- phase 1: load 70x38 halo tile (async DMA into LDS where available) ----
    for (int i = tid; i < RAW_H * RAW_W; i += 256) {
        const int r = i / RAW_W, c = i % RAW_W;
        const int gy = gy0 - RAD + r;
        const int gx = gx0 - RAD + c;
        const bool inb = (gy >= 0) & (gy < IMG_H) & (gx >= 0) & (gx < IMG_W);
        if (inb) {
            const float* gp = plane + (size_t)gy * IMG_W + gx;
#if USE_ASYNC
            __builtin_amdgcn_global_load_async_to_lds_b32(
                (int*)(void*)gp, (int*)&raw[r * RAW_P + c],
                /*offset=*/0, /*cpol=*/0);
#else
            raw[r * RAW_P + c] = *gp;
#endif
        }
    }
#if USE_ASYNC
    __builtin_amdgcn_s_wait_asynccnt(0);
#endif
    __syncthreads();

    // ---- phase 2: horizontal 7-tap sums of data and mask ----
    for (int i = tid; i < RAW_H * TILE_W; i += 256) {
        const int r = i / TILE_W, c = i % TILE_W;
        float s = 0.0f, m = 0.0f;
#pragma unroll
        for (int d = 0; d < 7; ++d) {
            const float v = raw[r * RAW_P + c + d];
            s += v;
            m += (v != 0.0f) ? 1.0f : 0.0f;
        }
        hsD[r * HS_P + c] = s;
        hsM[r * HS_P + c] = m;
    }
    __syncthreads();

    // ---- phase 3: vertical 7-tap sum as banded-ones matmul, one 16x16 tile/wave ----
    const int lane = tid & 31;
    const int wave = tid >> 5;
    const int wc0  = (wave & 3) << 4;   // 0,16,32,48
    const int wr0  = (wave >> 2) << 4;  // 0,16
    const int Mrow = lane & 15;
    const int hl   = lane >> 4;         // half-wave: 0 or 1
    const int cb   = wc0 + Mrow;        // B/C column (N = lane%16)

    v8f cs = {0.f,0.f,0.f,0.f,0.f,0.f,0.f,0.f};
    v8f cm = {0.f,0.f,0.f,0.f,0.f,0.f,0.f,0.f};

#if USE_WMMA
#pragma unroll
    for (int k0 = 0; k0 < 24; k0 += 4) {
        const int ka = k0 + 2 * hl;     // this lane's K for VGPR0 (A) / B row (VGPR0)
        v2f a, bS, bM;
        // Band matrix: A[M][K] = 1 iff M <= K <= M+6 (zero for K=22,23 pad)
        a.x = (ka     >= Mrow && ka     <= Mrow + 6) ? 1.0f : 0.0f;
        a.y = (ka + 1 >= Mrow && ka + 1 <= Mrow + 6) ? 1.0f : 0.0f;
        const int rb = wr0 + ka;        // hsum row feeding B VGPR0 (VGPR1 = rb+1)
        bS.x = hsD[rb * HS_P + cb];  bS.y = hsD[(rb + 1) * HS_P + cb];
        bM.x = hsM[rb * HS_P + cb];  bM.y = hsM[(rb + 1) * HS_P + cb];
        cs = __builtin_amdgcn_wmma_f32_16x16x4_f32(false, a, false, bS,
                                                   (short)0, cs, false, false);
        cm = __builtin_amdgcn_wmma_f32_16x16x4_f32(false, a, false, bM,
                                                   (short)0, cm, false, false);
    }
#else
#pragma unroll
    for (int j = 0; j < 8; ++j) {
        const int orow = wr0 + j + 8 * hl;
        float s = 0.0f, m = 0.0f;
#pragma unroll
        for (int d = 0; d < 7; ++d) {
            s += hsD[(orow + d) * HS_P + cb];
            m += hsM[(orow + d) * HS_P + cb];
        }
        cs[j] = s; cm[j] = m;
    }
#endif

    // ---- phase 4: combine (freeze filled pixels, avg for holes) + store ----
    // C/D layout: VGPR j -> M = j (lanes 0-15) / j+8 (lanes 16-31), N = lane%16
#pragma unroll
    for (int j = 0; j < 8; ++j) {
        const int orow = wr0 + j + 8 * hl;
        const float center = raw[(orow + RAD) * RAW_P + (cb + RAD)];
        const float cnt = cm[j];
        const float avg = (cnt > 0.0f) ? (cs[j] / cnt) : 0.0f;
        const float outv = (center != 0.0f) ? center : avg;
        oplane[(size_t)(gy0 + orow) * IMG_W + (gx0 + cb)] = outv;
    }
}

extern "C" void kernel_launch(void* const* d_in, const int* in_sizes, int n_in,
                              void* d_out, int out_size, void* d_ws, size_t ws_size,
                              hipStream_t stream) {
    (void)in_sizes; (void)n_in; (void)out_size; (void)ws_size;
    const float* src = (const float*)d_in[0];
    float* out = (float*)d_out;
    float* ws  = (float*)d_ws;   // needs BATCH*IMG_H*IMG_W*4 = 32 MB

    dim3 grid(IMG_W / TILE_W, IMG_H / TILE_H, BATCH);
    dim3 block(256);

    // Ping-pong: in -> ws -> out -> ws -> ... (ITERS=50 even => final in d_out)
    const float* cur = src;
    for (int i = 0; i < ITERS; ++i) {
        float* dst = (i & 1) ? out : ws;
        fill_iter_kernel<<<grid, block, 0, stream>>>(cur, dst);
        cur = dst;
    }
}